// TEMixtralSparseMoeBlock_70480413327791
// MI455X (gfx1250) — compile-verified
//
#include <hip/hip_runtime.h>
#include <hip/hip_bf16.h>
#include <math.h>

// ---------------- problem dims ----------------
#define HIDDEN 1024
#define FFN    3584
#define GUP    (2 * FFN)   // 7168
#define NEXP   8
#define NTOK   1024        // B*S = 1*1024

// LDS strides padded by 4 floats to spread the 64 banks
#define XS_STRIDE  (HIDDEN + 4)   // 1028
#define ACT_STRIDE (FFN + 4)      // 3588

typedef float v2f __attribute__((ext_vector_type(2)));
typedef float v8f __attribute__((ext_vector_type(8)));

// fp32 WMMA: D(16x16) = A(16x4) * B(4x16) + C, wave32.
// 8-arg pattern per probe: (neg_a, A, neg_b, B, c_mod, C, reuse_a, reuse_b)
__device__ __forceinline__ v8f wmma_f32(v2f a, v2f b, v8f c) {
    return __builtin_amdgcn_wmma_f32_16x16x4_f32(
        /*neg_a=*/false, a, /*neg_b=*/false, b,
        /*c_mod=*/(short)0, c, /*reuse_a=*/false, /*reuse_b=*/false);
}

// ---------------- kernel 0: zero output + expert counters ----------------
__global__ void moe_zero_kernel(float* __restrict__ out, int* __restrict__ cnt) {
    int gid = blockIdx.x * blockDim.x + threadIdx.x;
    if (gid < NTOK * HIDDEN) out[gid] = 0.0f;
    if (gid < NEXP) cnt[gid] = 0;
}

// ---------------- kernel 1: router (logits, softmax, top-2, dispatch) ----
// One wave (32 lanes) per token; 8 tokens per 256-thread block.
__global__ void moe_router_kernel(const float* __restrict__ x,
                                  const float* __restrict__ gw,
                                  float* __restrict__ logits_out,
                                  int*   __restrict__ cnt,
                                  int*   __restrict__ tok_idx,
                                  float* __restrict__ tok_w) {
    const int wave = threadIdx.x >> 5;
    const int lane = threadIdx.x & 31;
    const int t = blockIdx.x * 8 + wave;
    if (t >= NTOK) return;

    float acc[NEXP];
#pragma unroll
    for (int e = 0; e < NEXP; ++e) acc[e] = 0.0f;

    for (int h = lane; h < HIDDEN; h += 32) {
        float xv = x[t * HIDDEN + h];
#pragma unroll
        for (int e = 0; e < NEXP; ++e) acc[e] += xv * gw[e * HIDDEN + h];
    }

    // wave32 butterfly reduction
#pragma unroll
    for (int e = 0; e < NEXP; ++e) {
#pragma unroll
        for (int off = 16; off >= 1; off >>= 1)
            acc[e] += __shfl_xor(acc[e], off, 32);
    }

    if (lane == 0) {
        float m = acc[0];
#pragma unroll
        for (int e = 1; e < NEXP; ++e) m = fmaxf(m, acc[e]);
        float p[NEXP];
        float s = 0.0f;
#pragma unroll
        for (int e = 0; e < NEXP; ++e) { p[e] = __expf(acc[e] - m); s += p[e]; }
        float inv = 1.0f / s;
#pragma unroll
        for (int e = 0; e < NEXP; ++e) {
            p[e] *= inv;
            logits_out[t * NEXP + e] = acc[e];   // second tuple output
        }
        // top-2, first-index wins ties (matches jax.lax.top_k)
        int i1 = 0;
#pragma unroll
        for (int e = 1; e < NEXP; ++e) if (p[e] > p[i1]) i1 = e;
        int i2 = (i1 == 0) ? 1 : 0;
#pragma unroll
        for (int e = 0; e < NEXP; ++e) if (e != i1 && p[e] > p[i2]) i2 = e;
        float s2 = p[i1] + p[i2];
        float w1 = p[i1] / s2;
        float w2 = p[i2] / s2;

        int slot1 = atomicAdd(&cnt[i1], 1);
        tok_idx[i1 * NTOK + slot1] = t;
        tok_w  [i1 * NTOK + slot1] = w1;
        int slot2 = atomicAdd(&cnt[i2], 1);
        tok_idx[i2 * NTOK + slot2] = t;
        tok_w  [i2 * NTOK + slot2] = w2;
    }
}

// ---------------- kernel 2: fused per-expert tile (gate/up+SiLU+down) ----
// Grid: (64 token-tiles, 8 experts). Block: 256 threads = 8 waves.
// Dynamic LDS: x-tile 16x1028 + act-tile 16x3588 (~289 KB < 320 KB WGP LDS).
__global__ void moe_expert_kernel(const float* __restrict__ x,
                                  const float* __restrict__ w_gu,
                                  const float* __restrict__ w_dn,
                                  const int*   __restrict__ cnt,
                                  const int*   __restrict__ tok_idx,
                                  const float* __restrict__ tok_w,
                                  float* __restrict__ out) {
    const int e    = blockIdx.y;
    const int tile = blockIdx.x;
    const int c    = cnt[e];
    if (tile * 16 >= c) return;

    __shared__ int   s_tok[16];
    __shared__ float s_w[16];
    extern __shared__ float smem[];
    float* xs  = smem;                       // [16][XS_STRIDE]
    float* act = smem + 16 * XS_STRIDE;      // [16][ACT_STRIDE]

    if (threadIdx.x < 16) {
        int r = tile * 16 + threadIdx.x;
        if (r < c) {
            s_tok[threadIdx.x] = tok_idx[e * NTOK + r];
            s_w  [threadIdx.x] = tok_w  [e * NTOK + r];
        } else {       // pad rows: duplicate token 0 with weight 0 -> adds 0
            s_tok[threadIdx.x] = 0;
            s_w  [threadIdx.x] = 0.0f;
        }
    }
    __syncthreads();

    // Stage x rows into LDS (coalesced along H)
    for (int i = threadIdx.x; i < 16 * HIDDEN; i += blockDim.x) {
        int r = i >> 10;            // /HIDDEN
        int h = i & (HIDDEN - 1);
        xs[r * XS_STRIDE + h] = x[(size_t)s_tok[r] * HIDDEN + h];
    }
    __syncthreads();

    const int wave = threadIdx.x >> 5;
    const int lane = threadIdx.x & 31;
    const int half = lane >> 4;     // 0: K={k,k+1}, 1: K={k+2,k+3}
    const int l15  = lane & 15;

    // ---- Stage 1: gate/up projection + SiLU -> act in LDS ----
    const float* wg = w_gu + (size_t)e * HIDDEN * GUP;
    for (int nt = wave; nt < FFN / 16; nt += 8) {
        const int f0 = nt * 16;
        v8f ag = {};
        v8f au = {};
        const float* Bg = wg + f0 + l15;          // gate columns
        const float* Bu = wg + FFN + f0 + l15;    // up   columns
        const float* Ar = xs + l15 * XS_STRIDE + 2 * half;
        for (int k = 0; k < HIDDEN; k += 4) {
            v2f a;
            a.x = Ar[k];
            a.y = Ar[k + 1];
            const int kk = k + 2 * half;
            v2f bg, bu;
            bg.x = Bg[(size_t)kk * GUP];
            bg.y = Bg[(size_t)(kk + 1) * GUP];
            bu.x = Bu[(size_t)kk * GUP];
            bu.y = Bu[(size_t)(kk + 1) * GUP];
            ag = wmma_f32(a, bg, ag);
            au = wmma_f32(a, bu, au);
        }
#pragma unroll
        for (int v = 0; v < 8; ++v) {
            int M = v + 8 * half;
            float g = ag[v];
            float u = au[v];
            float sg = g / (1.0f + __expf(-g));   // SiLU
            act[M * ACT_STRIDE + f0 + l15] = sg * u;
        }
    }
    __syncthreads();

    // ---- Stage 2: down projection + weighted scatter-add ----
    const float* wd = w_dn + (size_t)e * FFN * HIDDEN;
    for (int nt = wave; nt < HIDDEN / 16; nt += 8) {
        const int h0 = nt * 16;
        v8f accd = {};
        const float* Bd = wd + h0 + l15;
        const float* Ar = act + l15 * ACT_STRIDE + 2 * half;
        for (int k = 0; k < FFN; k += 4) {
            v2f a;
            a.x = Ar[k];
            a.y = Ar[k + 1];
            const int kk = k + 2 * half;
            v2f b;
            b.x = Bd[(size_t)kk * HIDDEN];
            b.y = Bd[(size_t)(kk + 1) * HIDDEN];
            accd = wmma_f32(a, b, accd);
        }
#pragma unroll
        for (int v = 0; v < 8; ++v) {
            int M = v + 8 * half;
            float wv = s_w[M];
            atomicAdd(&out[(size_t)s_tok[M] * HIDDEN + h0 + l15], wv * accd[v]);
        }
    }
}

// ---------------- host launcher ----------------
extern "C" void kernel_launch(void* const* d_in, const int* in_sizes, int n_in,
                              void* d_out, int out_size, void* d_ws, size_t ws_size,
                              hipStream_t stream) {
    const float* x   = (const float*)d_in[0];   // [1,1024,1024]
    const float* gw  = (const float*)d_in[1];   // [8,1024]
    const float* wgu = (const float*)d_in[2];   // [8,1024,7168]
    const float* wdn = (const float*)d_in[3];   // [8,3584,1024]

    float* out    = (float*)d_out;              // [1024*1024] then logits
    float* logits = out + (size_t)NTOK * HIDDEN; // [1024*8]

    // workspace: counters, token lists, weights (~64 KB)
    int*   cnt  = (int*)d_ws;
    int*   tidx = cnt + NEXP;
    float* tw   = (float*)(tidx + NEXP * NTOK);

    moe_zero_kernel<<<(NTOK * HIDDEN + 255) / 256, 256, 0, stream>>>(out, cnt);
    moe_router_kernel<<<NTOK / 8, 256, 0, stream>>>(x, gw, logits, cnt, tidx, tw);

    const size_t smem = (size_t)(16 * XS_STRIDE + 16 * ACT_STRIDE) * sizeof(float);
    moe_expert_kernel<<<dim3(NTOK / 16, NEXP), 256, smem, stream>>>(
        x, wgu, wdn, cnt, tidx, tw, out);
}